// GATModule_86165633892568
// MI455X (gfx1250) — compile-verified
//
#include <hip/hip_runtime.h>

typedef float v2f __attribute__((ext_vector_type(2)));
typedef float v8f __attribute__((ext_vector_type(8)));
typedef unsigned int u32x4 __attribute__((ext_vector_type(4)));
typedef int i32x4 __attribute__((ext_vector_type(4)));
typedef int i32x8 __attribute__((ext_vector_type(8)));

#define EMB 256
#define NEG_SLOPE 0.2f

// ---------- helpers: order-preserving float<->uint for atomicMax ----------
__device__ __forceinline__ unsigned fenc(float f) {
    unsigned u = __float_as_uint(f);
    return (u & 0x80000000u) ? ~u : (u | 0x80000000u);
}
__device__ __forceinline__ float fdec(unsigned u) {
    return __uint_as_float((u & 0x80000000u) ? (u ^ 0x80000000u) : ~u);
}

// ---------- zero fill ----------
__global__ void k_zero(float* __restrict__ p, long long n) {
    long long i = (long long)blockIdx.x * blockDim.x + threadIdx.x;
    long long stride = (long long)gridDim.x * blockDim.x;
    for (; i < n; i += stride) p[i] = 0.0f;
}

// ---------- fp32 WMMA GEMM: Hout[N,256] = X[N,256] * W[256,256] ----------
// Block = 8 waves stacked along M, one 16-wide N strip per block. The shared
// B panel W[0:256, n0:n0+16] (16KB) is DMA'd into LDS once per block by the
// Tensor Data Mover (2D tile: 256 rows x 64B, stride 256 elems), then each
// wave runs 64 x V_WMMA_F32_16X16X4_F32 reading B fragments from LDS.
__global__ void __launch_bounds__(256)
k_gemm256_wmma(const float* __restrict__ X, const float* __restrict__ W,
               float* __restrict__ Hout, int N) {
    __shared__ float Bs[EMB * 16];          // B panel, row-major, 16 floats/row

    const int wave = threadIdx.x >> 5;
    const int lane = threadIdx.x & 31;
    const int n0 = blockIdx.y << 4;

#if __has_builtin(__builtin_amdgcn_tensor_load_to_lds) && __has_builtin(__builtin_amdgcn_s_wait_tensorcnt)
    if (wave == 0) {
        const unsigned lds_off = (unsigned)(unsigned long long)(Bs);      // LDS byte offset
        const unsigned long long ga = (unsigned long long)(const void*)(W + n0);
        u32x4 g0;
        g0.x = 1u;                                              // count=1 (valid), load
        g0.y = lds_off;                                         // lds_addr
        g0.z = (unsigned)ga;                                    // global_addr[31:0]
        g0.w = ((unsigned)(ga >> 32) & 0x01FFFFFFu)             // global_addr[56:32]
               | 0x80000000u;                                   // type=2 ("image")
        i32x8 g1;
        g1[0] = (int)0x00020000u;        // workgroup_mask=0, data_size=2 (4 bytes)
        g1[1] = (int)(16u << 16);        // tensor_dim0 = 16 elems (tile row width)
        g1[2] = (int)(256u << 16);       // tensor_dim0[31:16]=0 | tensor_dim1 = 256
        g1[3] = (int)(16u << 16);        // tensor_dim1[31:16]=0 | tile_dim0 = 16
        g1[4] = (int)256u;               // tile_dim1 = 256, tile_dim2 = 0 (2D)
        g1[5] = (int)256u;               // tensor_dim0_stride = 256 elems (row pitch)
        g1[6] = 0;                       // stride[47:32]=0, tensor_dim1_stride lo = 0
        g1[7] = 0;
        i32x4 g2 = {0, 0, 0, 0};         // unused for 2D tensor
        i32x4 g3 = {0, 0, 0, 0};
        i32x8 g4 = {0, 0, 0, 0, 0, 0, 0, 0};   // extra group (6-arg builtin form)
        __builtin_amdgcn_tensor_load_to_lds(g0, g1, g2, g3, g4, /*cpol=*/0);
        __builtin_amdgcn_s_wait_tensorcnt(0);
    }
#else
    for (int i = threadIdx.x; i < EMB * 16; i += 256) {
        const int row = i >> 4, col = i & 15;
        Bs[i] = W[(size_t)row * EMB + n0 + col];
    }
#endif
    __syncthreads();                     // B panel visible to all 8 waves

    const int m_tiles = (N + 15) >> 4;
    const int mt = blockIdx.x * 8 + wave;
    if (mt < m_tiles) {                  // wave-uniform guard (EXEC all-1 for WMMA)
        const int m0 = mt << 4;
        const int half = lane >> 4;      // 0: K pair {0,1}, 1: K pair {2,3}
        const int r    = lane & 15;      // M index (A) / N index (B)

        const float* __restrict__ xrow = X + (size_t)(m0 + r) * EMB;
        v8f acc = {0.f, 0.f, 0.f, 0.f, 0.f, 0.f, 0.f, 0.f};

        for (int k0 = 0; k0 < EMB; k0 += 4) {
            const int k = k0 + half * 2;
            // A 16x4 fp32 fragment: lane r holds row m0+r, K = k, k+1 (contiguous)
            v2f a = { xrow[k], xrow[k + 1] };
            // B 4x16 fp32 fragment from LDS: lane r holds col r, K = k, k+1
            v2f b = { Bs[k * 16 + r], Bs[(k + 1) * 16 + r] };
            acc = __builtin_amdgcn_wmma_f32_16x16x4_f32(
                /*neg_a=*/false, a, /*neg_b=*/false, b,
                /*c_mod=*/(short)0, acc, /*reuse_a=*/false, /*reuse_b=*/false);
        }

        // C layout: VGPR v, lanes 0-15 -> M = v, lanes 16-31 -> M = v+8, N = r
#pragma unroll
        for (int v = 0; v < 8; ++v) {
            const int m = m0 + v + 8 * half;
            Hout[(size_t)m * EMB + n0 + r] = acc[v];
        }
    }
}

// ---------- per-(node, head) attention scores ----------
__global__ void k_attn(const float* __restrict__ Hm,
                       const float* __restrict__ a_src, const float* __restrict__ a_dst,
                       float* __restrict__ al_s, float* __restrict__ al_d,
                       int N, int Hh, int F) {
    int i = blockIdx.x * blockDim.x + threadIdx.x;
    if (i >= N * Hh) return;
    const int n = i / Hh, h = i % Hh;
    const float* hp = Hm + (size_t)n * EMB + h * F;
    const float* as = a_src + h * F;
    const float* ad = a_dst + h * F;
    float s = 0.f, d = 0.f;
    for (int f = 0; f < F; ++f) { float v = hp[f]; s = fmaf(v, as[f], s); d = fmaf(v, ad[f], d); }
    al_s[i] = s; al_d[i] = d;
}

// ---------- per-(edge, head) logits + segment max ----------
__global__ void k_logits(const int* __restrict__ src, const int* __restrict__ dst,
                         int E, int N,
                         const float* __restrict__ al_s, const float* __restrict__ al_d,
                         int Hh, float* __restrict__ logits, unsigned* __restrict__ menc) {
    long long t = (long long)blockIdx.x * blockDim.x + threadIdx.x;
    long long total = (long long)(E + N) * Hh;
    if (t >= total) return;
    const int h = (int)(t % Hh);
    const long long e = t / Hh;
    const int s = (e < E) ? src[e] : (int)(e - E);   // self-loops appended
    const int d = (e < E) ? dst[e] : (int)(e - E);
    float x = al_s[(size_t)s * Hh + h] + al_d[(size_t)d * Hh + h];
    x = (x > 0.f) ? x : NEG_SLOPE * x;               // leaky_relu
    logits[t] = x;
    atomicMax(&menc[(size_t)d * Hh + h], fenc(x));
}

// ---------- per-(edge, head) exp + segment sum ----------
__global__ void k_expsum(const int* __restrict__ src, const int* __restrict__ dst,
                         int E, int N, int Hh,
                         float* __restrict__ logits, const unsigned* __restrict__ menc,
                         float* __restrict__ z) {
    long long t = (long long)blockIdx.x * blockDim.x + threadIdx.x;
    long long total = (long long)(E + N) * Hh;
    if (t >= total) return;
    const int h = (int)(t % Hh);
    const long long e = t / Hh;
    const int d = (e < E) ? dst[e] : (int)(e - E);
    const float m = fdec(menc[(size_t)d * Hh + h]);
    const float ev = expf(logits[t] - m);
    logits[t] = ev;                                  // reuse buffer for numerators
    atomicAdd(&z[(size_t)d * Hh + h], ev);
}

// ---------- per-(edge, float4) weighted scatter-add ----------
__global__ void k_agg(const int* __restrict__ src, const int* __restrict__ dst,
                      int E, int N, int Hh, int F,
                      const float* __restrict__ Hm, const float* __restrict__ ev,
                      const float* __restrict__ z, float* __restrict__ out) {
    long long t = (long long)blockIdx.x * blockDim.x + threadIdx.x;
    long long total = (long long)(E + N) * 64;       // 64 float4 per edge (256 floats)
    if (t >= total) return;
    const long long e = t >> 6;
    const int c = ((int)(t & 63)) << 2;              // column 0..252 step 4
    const int h = c / F;
    const int s = (e < E) ? src[e] : (int)(e - E);
    const int d = (e < E) ? dst[e] : (int)(e - E);
    const float alpha = ev[(size_t)e * Hh + h] / (z[(size_t)d * Hh + h] + 1e-16f);
    const float4 hv = *(const float4*)(Hm + (size_t)s * EMB + c);
    float* op = out + (size_t)d * EMB + c;
    atomicAdd(op + 0, hv.x * alpha);
    atomicAdd(op + 1, hv.y * alpha);
    atomicAdd(op + 2, hv.z * alpha);
    atomicAdd(op + 3, hv.w * alpha);
}

// ---------- bias add ----------
__global__ void k_bias(float* __restrict__ out, const float* __restrict__ b, long long total) {
    long long i = (long long)blockIdx.x * blockDim.x + threadIdx.x;
    if (i < total) out[i] += b[i % EMB];
}

extern "C" void kernel_launch(void* const* d_in, const int* in_sizes, int n_in,
                              void* d_out, int out_size, void* d_ws, size_t ws_size,
                              hipStream_t stream) {
    const float* x   = (const float*)d_in[0];
    const int*   ei  = (const int*)d_in[1];    // [2, E] int32 (harness int convention)
    const float* W1  = (const float*)d_in[2];  // [256, 8, 32] -> flat [256,256]
    const float* as1 = (const float*)d_in[3];
    const float* ad1 = (const float*)d_in[4];
    const float* b1  = (const float*)d_in[5];
    const float* W2  = (const float*)d_in[6];  // [256, 1, 256] -> flat [256,256]
    const float* as2 = (const float*)d_in[7];
    const float* ad2 = (const float*)d_in[8];
    const float* b2  = (const float*)d_in[9];

    const int N = in_sizes[0] / EMB;           // 50000
    const int E = in_sizes[1] / 2;             // 800000
    const int* src = ei;
    const int* dst = ei + E;
    const long long ET = (long long)E + N;     // with self-loops

    // workspace layout (floats)
    float*    hbuf   = (float*)d_ws;                       // N*EMB
    float*    out1   = hbuf + (size_t)N * EMB;             // N*EMB
    float*    al_s   = out1 + (size_t)N * EMB;             // N*8 (max H)
    float*    al_d   = al_s + (size_t)N * 8;               // N*8
    unsigned* menc   = (unsigned*)(al_d + (size_t)N * 8);  // N*8
    float*    zbuf   = (float*)(menc + (size_t)N * 8);     // N*8
    float*    logits = zbuf + (size_t)N * 8;               // ET*8
    (void)ws_size; (void)n_in; (void)out_size;

    auto layer = [&](const float* Xin, const float* Wm, const float* as,
                     const float* ad, const float* bb, float* Yout, int Hh) {
        const int F = EMB / Hh;
        const long long nelems = (long long)N * EMB;
        const long long nh     = (long long)N * Hh;
        const long long nt     = ET * Hh;
        const long long at     = ET * 64;

        k_zero<<<(unsigned)((nelems + 255) / 256), 256, 0, stream>>>(Yout, nelems);
        k_zero<<<(unsigned)((nh + 255) / 256), 256, 0, stream>>>((float*)menc, nh);
        k_zero<<<(unsigned)((nh + 255) / 256), 256, 0, stream>>>(zbuf, nh);

        const int m_tiles = (N + 15) / 16;
        dim3 gg((unsigned)((m_tiles + 7) / 8), EMB / 16);
        k_gemm256_wmma<<<gg, 256, 0, stream>>>(Xin, Wm, hbuf, N);

        k_attn<<<(unsigned)((N * Hh + 255) / 256), 256, 0, stream>>>(
            hbuf, as, ad, al_s, al_d, N, Hh, F);
        k_logits<<<(unsigned)((nt + 255) / 256), 256, 0, stream>>>(
            src, dst, E, N, al_s, al_d, Hh, logits, menc);
        k_expsum<<<(unsigned)((nt + 255) / 256), 256, 0, stream>>>(
            src, dst, E, N, Hh, logits, menc, zbuf);
        k_agg<<<(unsigned)((at + 255) / 256), 256, 0, stream>>>(
            src, dst, E, N, Hh, F, hbuf, logits, zbuf, Yout);
        k_bias<<<(unsigned)((nelems + 255) / 256), 256, 0, stream>>>(Yout, bb, nelems);
    };

    layer(x,    W1, as1, ad1, b1, out1,           8);   // layer 1: H=8, F=32
    layer(out1, W2, as2, ad2, b2, (float*)d_out,  1);   // layer 2: H=1, F=256
}